// GCN300_51488067944595
// MI455X (gfx1250) — compile-verified
//
#include <hip/hip_runtime.h>

typedef __attribute__((ext_vector_type(2))) float v2f;
typedef __attribute__((ext_vector_type(8))) float v8f;

#define N_NODES 300000
#define N_EDGES 3000000
#define M_TILES 18750   // N_NODES / 16
#define BN_EPS  1e-5f

// ---------------------------------------------------------------------------
// Graph normalization: deg (with self loop), dinv = rsqrt(deg), norm per edge
// ---------------------------------------------------------------------------
__global__ void deg_init_kernel(float* deg, int n) {
    int i = blockIdx.x * blockDim.x + threadIdx.x;
    if (i < n) deg[i] = 1.0f;   // self loop contributes 1
}

__global__ void deg_acc_kernel(const long long* __restrict__ dst, float* deg, int e) {
    int i = blockIdx.x * blockDim.x + threadIdx.x;
    if (i < e) atomicAdd(&deg[(int)dst[i]], 1.0f);
}

__global__ void rsqrt_ip_kernel(float* d, int n) {
    int i = blockIdx.x * blockDim.x + threadIdx.x;
    if (i < n) d[i] = rsqrtf(d[i]);   // deg >= 1 always
}

__global__ void norm_kernel(const long long* __restrict__ src,
                            const long long* __restrict__ dst,
                            const float* __restrict__ dinv,
                            float* __restrict__ norm, int e) {
    int i = blockIdx.x * blockDim.x + threadIdx.x;
    if (i < e) norm[i] = dinv[(int)src[i]] * dinv[(int)dst[i]];
}

// ---------------------------------------------------------------------------
// Weight pre-packing into WMMA B-fragment layout, zero padded in K and N.
// Bfrag[((s*2+half)*NW + col)*2 + j] = W[4s + 2*half + j][col]  (0 if OOB)
// Flat index i: j = i&1, col = (i>>1)%NW, sh = (i>>1)/NW, k = 2*sh + j.
// ---------------------------------------------------------------------------
__global__ void pack_bfrag_kernel(const float* __restrict__ W, float* __restrict__ Wp,
                                  int K, int Ncols, int NW, int total) {
    int i = blockIdx.x * blockDim.x + threadIdx.x;
    if (i >= total) return;
    const int j   = i & 1;
    const int r   = i >> 1;
    const int col = r % NW;
    const int kk  = 2 * (r / NW) + j;
    Wp[i] = (kk < K && col < Ncols) ? W[kk * Ncols + col] : 0.f;
}

// x (N x 25) -> xpad (N x 28), zero padded columns
__global__ void pack_x_kernel(const float* __restrict__ x, float* __restrict__ xp, int n) {
    int i = blockIdx.x * blockDim.x + threadIdx.x;   // n = N*28
    if (i >= n) return;
    const int k = i % 28;
    xp[i] = (k < 25) ? x[(i / 28) * 25 + k] : 0.f;
}

// ---------------------------------------------------------------------------
// Fused MLP: h0 = BN2( relu(BN1(x@W1 + b1)) @ W2 + b2 )  -> stride-28 padded
// Stage-1: xpad(16x28) @ W1frag (7 col tiles of 16) -> BN1 -> LDS (stride 114)
// Stage-2: relu(LDS)(16x100) @ W2frag (2 col tiles) -> BN2 -> h0
// All fragment loads are unconditional aligned 8-byte loads.
// ---------------------------------------------------------------------------
__global__ __launch_bounds__(192) void ffn1_kernel(
    const float* __restrict__ xpad,
    const float* __restrict__ W1f,
    const float* __restrict__ b1, const float* __restrict__ g1,
    const float* __restrict__ be1, const float* __restrict__ m1,
    const float* __restrict__ v1,
    const float* __restrict__ W2f,
    const float* __restrict__ b2, const float* __restrict__ g2,
    const float* __restrict__ be2, const float* __restrict__ m2,
    const float* __restrict__ v2,
    float* __restrict__ h0)
{
    const int lane  = threadIdx.x;          // 0..31
    const int wid   = threadIdx.y;          // 0..5
    const int tileM = blockIdx.x * 6 + wid; // 0..18749 (exact)
    const int half  = lane >> 4;
    const int l16   = lane & 15;
    const int row0  = tileM * 16;

    __shared__ float lds[6][16][114];       // even stride: aligned v2f reads
    float (*T)[114] = lds[wid];

    const float* Ar = xpad + (size_t)(row0 + l16) * 28 + 2 * half;

    // ---- stage 1: 25->100 (K pad 28, 7 col tiles of 16, NW=112) ----
    for (int tn = 0; tn < 7; ++tn) {
        const int    col = tn * 16 + l16;
        const float* Bf  = W1f + 2 * (col + half * 112);
        v8f c = {};
        #pragma unroll
        for (int s = 0; s < 7; ++s) {
            v2f a = *(const v2f*)(Ar + 4 * s);
            v2f b = *(const v2f*)(Bf + 4 * 112 * s);
            c = __builtin_amdgcn_wmma_f32_16x16x4_f32(false, a, false, b,
                                                      (short)0, c, false, false);
        }
        float sc = 0.f, sh = 0.f;
        if (col < 100) {
            sc = g1[col] * rsqrtf(v1[col] + BN_EPS);
            sh = (b1[col] - m1[col]) * sc + be1[col];
        }
        #pragma unroll
        for (int r = 0; r < 8; ++r)
            T[r + 8 * half][col] = c[r] * sc + sh;   // cols >=100 land as 0
    }
    __syncthreads();

    // ---- stage 2: 100->25 (K=100, 2 col tiles, NW=32), relu fused on A ----
    const float* Lr = &T[l16][2 * half];
    for (int tn = 0; tn < 2; ++tn) {
        const int    col = tn * 16 + l16;
        const float* Bf  = W2f + 2 * (col + half * 32);
        v8f c = {};
        #pragma unroll
        for (int s = 0; s < 25; ++s) {
            v2f a = *(const v2f*)(Lr + 4 * s);
            a.x = fmaxf(a.x, 0.f); a.y = fmaxf(a.y, 0.f);
            v2f b = *(const v2f*)(Bf + 4 * 32 * s);
            c = __builtin_amdgcn_wmma_f32_16x16x4_f32(false, a, false, b,
                                                      (short)0, c, false, false);
        }
        float sc = 0.f, sh = 0.f;
        if (col < 25) {
            sc = g2[col] * rsqrtf(v2[col] + BN_EPS);
            sh = (b2[col] - m2[col]) * sc + be2[col];
        }
        if (col < 28) {                       // pad cols 25..27 get exact 0
            float* O = h0 + (size_t)(row0 + 8 * half) * 28 + col;
            #pragma unroll
            for (int r = 0; r < 8; ++r)
                O[r * 28] = c[r] * sc + sh;
        }
    }
}

// ---------------------------------------------------------------------------
// GCN dense node transform: Msg = act(H) @ W (B pre-packed, zero padded).
// One wave per (16-node, 16-col) tile. Unconditional aligned v2f loads.
// ---------------------------------------------------------------------------
template <int P_IN, int F_OUT, int P_OUT, int NCT, bool RELU_IN>
__global__ __launch_bounds__(192) void transform_kernel(
    const float* __restrict__ H, const float* __restrict__ Wf,
    float* __restrict__ Msg)
{
    constexpr int NW = NCT * 16;
    constexpr int KS = P_IN / 4;
    const int lane  = threadIdx.x;
    const int w     = blockIdx.x * 6 + threadIdx.y;
    const int tileM = (NCT == 1) ? w : (w / NCT);
    const int tn    = (NCT == 1) ? 0 : (w % NCT);
    const int half  = lane >> 4;
    const int l16   = lane & 15;
    const int row0  = tileM * 16;
    const int col   = tn * 16 + l16;

    const float* Ar = H  + (size_t)(row0 + l16) * P_IN + 2 * half;
    const float* Bf = Wf + 2 * (col + half * NW);

    v8f c = {};
    #pragma unroll
    for (int s = 0; s < KS; ++s) {
        v2f a = *(const v2f*)(Ar + 4 * s);
        if (RELU_IN) { a.x = fmaxf(a.x, 0.f); a.y = fmaxf(a.y, 0.f); }
        v2f b = *(const v2f*)(Bf + 4 * NW * s);
        c = __builtin_amdgcn_wmma_f32_16x16x4_f32(false, a, false, b,
                                                  (short)0, c, false, false);
    }
    if (col < P_OUT) {                 // pad cols (F_OUT..P_OUT) write exact 0
        float* O = Msg + (size_t)(row0 + 8 * half) * P_OUT + col;
        #pragma unroll
        for (int r = 0; r < 8; ++r)
            O[r * P_OUT] = c[r];
    }
}

// out[i,f] = bias[f] + dinv[i]^2 * msg[i,f]; pad columns stay exactly 0
template <int F, int P>
__global__ void agg_init_kernel(const float* __restrict__ msg,
                                const float* __restrict__ dinv,
                                const float* __restrict__ bias,
                                float* __restrict__ out, int n)
{
    int i = blockIdx.x * blockDim.x + threadIdx.x;
    if (i >= n * P) return;
    const int node = i / P, f = i % P;
    const float di = dinv[node];
    const float bb = (f < F) ? bias[f] : 0.f;
    out[i] = bb + di * di * msg[i];
}

// out[dst,:] += norm * msg[src,:]  (L2-resident float4 gather + f32 atomics)
template <int F, int P>
__global__ void scatter_kernel(const float* __restrict__ msg,
                               const long long* __restrict__ src,
                               const long long* __restrict__ dst,
                               const float* __restrict__ norm,
                               float* __restrict__ out, int e)
{
    int i = blockIdx.x * blockDim.x + threadIdx.x;
    if (i >= e) return;
    const int s = (int)src[i], d = (int)dst[i];
    const float wgt = norm[i];
    const float*  ms  = msg + (size_t)s * P;
    const float4* ms4 = (const float4*)ms;
    float* od = out + (size_t)d * P;
    constexpr int F4 = F / 4;
    #pragma unroll
    for (int q = 0; q < F4; ++q) {
        const float4 v = ms4[q];
        atomicAdd(&od[4 * q + 0], wgt * v.x);
        atomicAdd(&od[4 * q + 1], wgt * v.y);
        atomicAdd(&od[4 * q + 2], wgt * v.z);
        atomicAdd(&od[4 * q + 3], wgt * v.w);
    }
    #pragma unroll
    for (int f = F4 * 4; f < F; ++f)
        atomicAdd(&od[f], wgt * ms[f]);
}

// final FC: (1000 x 1200) @ (1200 x 4) + bfc, relu fused on load
__global__ void fc_kernel(const float* __restrict__ h,
                          const float* __restrict__ Wfc,
                          const float* __restrict__ bfc,
                          float* __restrict__ out)
{
    int t = blockIdx.x * blockDim.x + threadIdx.x;
    if (t >= 1000 * 4) return;
    const int r = t >> 2, c = t & 3;
    const float* hr = h + r * 1200;
    float acc = bfc[c];
    for (int k = 0; k < 1200; ++k)
        acc += fmaxf(hr[k], 0.f) * Wfc[k * 4 + c];
    out[t] = acc;
}

// ---------------------------------------------------------------------------
extern "C" void kernel_launch(void* const* d_in, const int* in_sizes, int n_in,
                              void* d_out, int out_size, void* d_ws, size_t ws_size,
                              hipStream_t stream)
{
    const float*     x   = (const float*)d_in[0];
    const long long* ei  = (const long long*)d_in[1];  // (2,E) int64 row-major
    const float *W1 = (const float*)d_in[2],  *b1  = (const float*)d_in[3];
    const float *g1 = (const float*)d_in[4],  *be1 = (const float*)d_in[5];
    const float *m1 = (const float*)d_in[6],  *v1  = (const float*)d_in[7];
    const float *W2 = (const float*)d_in[8],  *b2  = (const float*)d_in[9];
    const float *g2 = (const float*)d_in[10], *be2 = (const float*)d_in[11];
    const float *m2 = (const float*)d_in[12], *v2  = (const float*)d_in[13];
    const float *Wc1 = (const float*)d_in[14], *bc1 = (const float*)d_in[15];
    const float *Wc2 = (const float*)d_in[16], *bc2 = (const float*)d_in[17];
    const float *Wc3 = (const float*)d_in[18], *bc3 = (const float*)d_in[19];
    const float *Wc4 = (const float*)d_in[20], *bc4 = (const float*)d_in[21];
    const float *Wc5 = (const float*)d_in[22], *bc5 = (const float*)d_in[23];
    const float *Wfc = (const float*)d_in[24], *bfc = (const float*)d_in[25];

    const int N = N_NODES, E = N_EDGES;
    const long long* srcI = ei;
    const long long* dstI = ei + E;

    float* ws   = (float*)d_ws;
    float* dinv = ws;                              // N
    float* norm = dinv + N;                        // E
    float* xpad = norm + E;                        // N*28
    float* bufA = xpad + (size_t)N * 28;           // N*28 each
    float* bufB = bufA + (size_t)N * 28;
    float* bufC = bufB + (size_t)N * 28;
    float* W1f  = bufC + (size_t)N * 28;           // 28*112 = 3136
    float* W2f  = W1f + 3136;                      // 100*32 = 3200
    float* Wc1f = W2f + 3200;                      // 28*32  = 896
    float* Wc2f = Wc1f + 896;                      // 28*16  = 448
    float* Wc3f = Wc2f + 448;                      // 16*16  = 256
    float* Wc4f = Wc3f + 256;                      // 16*16  = 256
    float* Wc5f = Wc4f + 256;                      // 8*16   = 128

    const dim3 wblk(32, 6);                        // 6 wave32s / block
    const int T = 256;

    // ---- weight fragment packing + x padding ----
    pack_bfrag_kernel<<<(3136 + T - 1) / T, T, 0, stream>>>(W1,  W1f,  25, 100, 112, 3136);
    pack_bfrag_kernel<<<(3200 + T - 1) / T, T, 0, stream>>>(W2,  W2f, 100,  25,  32, 3200);
    pack_bfrag_kernel<<<(896  + T - 1) / T, T, 0, stream>>>(Wc1, Wc1f, 25,  25,  32, 896);
    pack_bfrag_kernel<<<(448  + T - 1) / T, T, 0, stream>>>(Wc2, Wc2f, 25,  16,  16, 448);
    pack_bfrag_kernel<<<(256  + T - 1) / T, T, 0, stream>>>(Wc3, Wc3f, 16,  16,  16, 256);
    pack_bfrag_kernel<<<(256  + T - 1) / T, T, 0, stream>>>(Wc4, Wc4f, 16,   8,  16, 256);
    pack_bfrag_kernel<<<(128  + T - 1) / T, T, 0, stream>>>(Wc5, Wc5f,  8,   4,  16, 128);
    pack_x_kernel<<<((size_t)N * 28 + T - 1) / T, T, 0, stream>>>(x, xpad, N * 28);

    // ---- graph normalization (computed once, reused by all 5 layers) ----
    deg_init_kernel<<<(N + T - 1) / T, T, 0, stream>>>(dinv, N);
    deg_acc_kernel<<<(E + T - 1) / T, T, 0, stream>>>(dstI, dinv, E);
    rsqrt_ip_kernel<<<(N + T - 1) / T, T, 0, stream>>>(dinv, N);
    norm_kernel<<<(E + T - 1) / T, T, 0, stream>>>(srcI, dstI, dinv, norm, E);

    // ---- fused MLP -> h0 (stride 28) in bufA ----
    ffn1_kernel<<<M_TILES / 6, wblk, 0, stream>>>(
        xpad, W1f, b1, g1, be1, m1, v1, W2f, b2, g2, be2, m2, v2, bufA);

    float* h = bufA; float* msg = bufB; float* agg = bufC;
    #define ROTATE() do { float* tmp = h; h = agg; agg = msg; msg = tmp; } while (0)

    // layer 1: 25->25 (no relu on input), strides 28->28, 2 col tiles
    transform_kernel<28, 25, 28, 2, false><<<M_TILES * 2 / 6, wblk, 0, stream>>>(h, Wc1f, msg);
    agg_init_kernel<25, 28><<<((size_t)N * 28 + T - 1) / T, T, 0, stream>>>(msg, dinv, bc1, agg, N);
    scatter_kernel<25, 28><<<(E + T - 1) / T, T, 0, stream>>>(msg, srcI, dstI, norm, agg, E);
    ROTATE();
    // layer 2: 25->16
    transform_kernel<28, 16, 16, 1, true><<<M_TILES / 6, wblk, 0, stream>>>(h, Wc2f, msg);
    agg_init_kernel<16, 16><<<((size_t)N * 16 + T - 1) / T, T, 0, stream>>>(msg, dinv, bc2, agg, N);
    scatter_kernel<16, 16><<<(E + T - 1) / T, T, 0, stream>>>(msg, srcI, dstI, norm, agg, E);
    ROTATE();
    // layer 3: 16->16
    transform_kernel<16, 16, 16, 1, true><<<M_TILES / 6, wblk, 0, stream>>>(h, Wc3f, msg);
    agg_init_kernel<16, 16><<<((size_t)N * 16 + T - 1) / T, T, 0, stream>>>(msg, dinv, bc3, agg, N);
    scatter_kernel<16, 16><<<(E + T - 1) / T, T, 0, stream>>>(msg, srcI, dstI, norm, agg, E);
    ROTATE();
    // layer 4: 16->8
    transform_kernel<16, 8, 8, 1, true><<<M_TILES / 6, wblk, 0, stream>>>(h, Wc4f, msg);
    agg_init_kernel<8, 8><<<((size_t)N * 8 + T - 1) / T, T, 0, stream>>>(msg, dinv, bc4, agg, N);
    scatter_kernel<8, 8><<<(E + T - 1) / T, T, 0, stream>>>(msg, srcI, dstI, norm, agg, E);
    ROTATE();
    // layer 5: 8->4
    transform_kernel<8, 4, 4, 1, true><<<M_TILES / 6, wblk, 0, stream>>>(h, Wc5f, msg);
    agg_init_kernel<4, 4><<<((size_t)N * 4 + T - 1) / T, T, 0, stream>>>(msg, dinv, bc5, agg, N);
    scatter_kernel<4, 4><<<(E + T - 1) / T, T, 0, stream>>>(msg, srcI, dstI, norm, agg, E);
    ROTATE();

    // final FC (relu fused on load), out is (1000,4) fp32
    fc_kernel<<<(4000 + T - 1) / T, T, 0, stream>>>(h, Wfc, bfc, (float*)d_out);
    #undef ROTATE
}